// Wav2VecLoss_49211735278067
// MI455X (gfx1250) — compile-verified
//
#include <hip/hip_runtime.h>

// Problem constants (from reference)
#define KSTEPS  12
#define NNEG    10
#define DDIM    256
#define BSZ     8
#define TLEN    1024

#define GEMM_BLOCKS  256    // 2048 waves / 8 waves per block
#define SCORE_BLOCKS 1024   // 8192 (b,t) waves / 8 waves per block

typedef float v2f __attribute__((ext_vector_type(2)));
typedef float v8f __attribute__((ext_vector_type(8)));

__device__ __forceinline__ float wave_sum(float v) {
#pragma unroll
    for (int off = 16; off > 0; off >>= 1) v += __shfl_xor(v, off, 32);
    return v;
}

// numerically stable log(sigmoid(x)) = min(x,0) - log1p(exp(-|x|))
__device__ __forceinline__ float log_sigmoid(float x) {
    return fminf(x, 0.0f) - log1pf(__expf(-fabsf(x)));
}

// Batched 32x32-tiled transpose: in [batch][rows][cols] -> out [batch][cols][rows]
__global__ void __launch_bounds__(256)
transpose_kernel(const float* __restrict__ in, float* __restrict__ out,
                 int rows, int cols) {
    __shared__ float tile[32][33];
    const int batch = blockIdx.z;
    const int c0 = blockIdx.x * 32, r0 = blockIdx.y * 32;
    const float* src = in  + (size_t)batch * rows * cols;
    float*       dst = out + (size_t)batch * rows * cols;
    const int tx = threadIdx.x & 31, ty = threadIdx.x >> 5;
#pragma unroll
    for (int j = 0; j < 4; ++j)
        tile[ty + j * 8][tx] = src[(size_t)(r0 + ty + j * 8) * cols + c0 + tx];
    __syncthreads();
#pragma unroll
    for (int j = 0; j < 4; ++j)
        dst[(size_t)(c0 + ty + j * 8) * rows + r0 + tx] = tile[tx][ty + j * 8];
}

// cst[b][t][e] = sum_d c[b][d][t] * Wt[d][e] + bias[e]
// WMMA f32 16x16x4: M = t-tile (16), N = e-tile (16), K = d (4 per op).
// Each wave: one 16(t) x 64(e) output strip, 4 accumulators, A reused 4x.
__global__ void __launch_bounds__(256)
gemm_kernel(const float* __restrict__ c, const float* __restrict__ Wt,
            const float* __restrict__ bias, float* __restrict__ cst) {
    const int lane = threadIdx.x & 31;
    const int wid  = threadIdx.x >> 5;
    const int wg   = blockIdx.x * 8 + wid;      // 0..2047
    const int eg   = wg & 3;                    // e-group (64 wide)
    const int tt   = (wg >> 2) & 63;            // t-tile
    const int b    = wg >> 8;                   // batch
    const int t0   = tt * 16;
    const int e0   = eg * 64;
    const int m    = lane & 15;
    const int jb   = (lane < 16) ? 0 : 2;       // K sub-offset per half-wave

    // A element (M=m, K=kk+j) = c[b][kk+j][t0+m]   (coalesced across lanes in t)
    const float* ap = c + (size_t)b * DDIM * TLEN + t0 + m;
    // B element (K=kk+j, N=m+16s) = Wt[kk+j][e0+16s+m] (coalesced across lanes in e)
    const float* bp = Wt + e0 + m;

    v8f acc0 = {}, acc1 = {}, acc2 = {}, acc3 = {};
#pragma unroll 4
    for (int kk = 0; kk < DDIM; kk += 4) {
        v2f a;
        a.x = ap[(size_t)(kk + jb) * TLEN];
        a.y = ap[(size_t)(kk + jb + 1) * TLEN];
        const float* bq = bp + (size_t)(kk + jb) * DDIM;
        v2f b0, b1, b2, b3;
        b0.x = bq[0];   b0.y = bq[DDIM];
        b1.x = bq[16];  b1.y = bq[DDIM + 16];
        b2.x = bq[32];  b2.y = bq[DDIM + 32];
        b3.x = bq[48];  b3.y = bq[DDIM + 48];
        acc0 = __builtin_amdgcn_wmma_f32_16x16x4_f32(false, a, false, b0, (short)0, acc0, false, false);
        acc1 = __builtin_amdgcn_wmma_f32_16x16x4_f32(false, a, false, b1, (short)0, acc1, false, false);
        acc2 = __builtin_amdgcn_wmma_f32_16x16x4_f32(false, a, false, b2, (short)0, acc2, false, false);
        acc3 = __builtin_amdgcn_wmma_f32_16x16x4_f32(false, a, false, b3, (short)0, acc3, false, false);
    }

    const float bv0 = bias[e0 + m];
    const float bv1 = bias[e0 + 16 + m];
    const float bv2 = bias[e0 + 32 + m];
    const float bv3 = bias[e0 + 48 + m];

    // C/D layout: VGPR r -> t-row (r for lanes 0-15, 8+r for lanes 16-31), N = lane%16
    const int thi = (lane < 16) ? 0 : 8;
    float* op = cst + ((size_t)b * TLEN + t0 + thi) * DDIM + e0 + m;
#pragma unroll
    for (int r = 0; r < 8; ++r) {
        op[(size_t)r * DDIM +  0] = acc0[r] + bv0;
        op[(size_t)r * DDIM + 16] = acc1[r] + bv1;
        op[(size_t)r * DDIM + 32] = acc2[r] + bv2;
        op[(size_t)r * DDIM + 48] = acc3[r] + bv3;
    }
}

// One wave per (b,t): positive dot + NNEG gathered negative dots over D=256.
// cst column and zt rows are contiguous (1 KB) -> float4 loads per lane.
__global__ void __launch_bounds__(256)
score_kernel(const float* __restrict__ zt, const float* __restrict__ cst,
             const int* __restrict__ negk, float* __restrict__ partials, int k) {
    __shared__ float sp[8], sn[8];
    const int lane = threadIdx.x & 31;
    const int wid  = threadIdx.x >> 5;
    const int gw   = blockIdx.x * 8 + wid;   // 0..8191
    const int b    = gw >> 10;
    const int t    = gw & (TLEN - 1);
    const int valid = (t < TLEN - k);

    const float* cp = cst + ((size_t)b * TLEN + t) * DDIM + lane * 8;
    const float4 c0 = *(const float4*)(cp);
    const float4 c1 = *(const float4*)(cp + 4);

    const float* zbase = zt + (size_t)b * TLEN * DDIM + lane * 8;

    auto dot_col = [&](int col) -> float {
        const float* zp = zbase + (size_t)col * DDIM;
        const float4 z0 = *(const float4*)(zp);
        const float4 z1 = *(const float4*)(zp + 4);
        float d = z0.x * c0.x + z0.y * c0.y + z0.z * c0.z + z0.w * c0.w
                + z1.x * c1.x + z1.y * c1.y + z1.z * c1.z + z1.w * c1.w;
        return wave_sum(d);
    };

    const int pcol = valid ? (t + k) : t;
    const float p = dot_col(pcol);
    float pos = valid ? log_sigmoid(p) : 0.0f;

    const int* ip = negk + ((size_t)b * TLEN + t) * NNEG;
    float neg = 0.0f;
#pragma unroll
    for (int n = 0; n < NNEG; ++n) {
        const int idx = ip[n];           // in [0, T-k)
        const float s = dot_col(idx);
        neg += log_sigmoid(-s);
    }
    if (!valid) neg = 0.0f;

    if (lane == 0) { sp[wid] = pos; sn[wid] = neg; }
    __syncthreads();
    if (threadIdx.x == 0) {
        float psum = 0.f, nsum = 0.f;
#pragma unroll
        for (int w = 0; w < 8; ++w) { psum += sp[w]; nsum += sn[w]; }
        partials[(size_t)blockIdx.x * 2 + 0] = psum;
        partials[(size_t)blockIdx.x * 2 + 1] = nsum;
    }
}

// Deterministic final reduction: fixed lane->entry assignment, then shuffle tree.
__global__ void reduce_kernel(const float* __restrict__ partials, float* __restrict__ out) {
    const int lane = threadIdx.x;
    float psum = 0.f, nsum = 0.f;
    const int total = KSTEPS * SCORE_BLOCKS;
    for (int i = lane; i < total; i += 32) {
        psum += partials[(size_t)i * 2 + 0];
        nsum += partials[(size_t)i * 2 + 1];
    }
    psum = wave_sum(psum);
    nsum = wave_sum(nsum);
    if (lane == 0) {
        out[0] = -psum;
        out[1] = -nsum;
        out[2] = -(psum + (float)NNEG * nsum);
    }
}

extern "C" void kernel_launch(void* const* d_in, const int* in_sizes, int n_in,
                              void* d_out, int out_size, void* d_ws, size_t ws_size,
                              hipStream_t stream) {
    (void)in_sizes; (void)n_in; (void)out_size; (void)ws_size;
    const float* z    = (const float*)d_in[0];   // [B][D][T]
    const float* c    = (const float*)d_in[1];   // [B][D][T]
    const float* W    = (const float*)d_in[2];   // [K][D][D] (e-major)
    const float* bias = (const float*)d_in[3];   // [K][D]
    const int*   neg  = (const int*)d_in[4];     // [K][B][T][NNEG]
    float* out = (float*)d_out;

    float* ws       = (float*)d_ws;
    float* Wt       = ws;                                  // [K][D][D] d-major (3 MB)
    float* zt       = Wt + (size_t)KSTEPS * DDIM * DDIM;   // [B][T][D]        (8 MB)
    float* cst      = zt + (size_t)BSZ * TLEN * DDIM;      // [B][T][D]        (8 MB)
    float* partials = cst + (size_t)BSZ * TLEN * DDIM;     // [K][1024][2]

    // Wt[k][d][e] = W[k][e][d]
    transpose_kernel<<<dim3(DDIM / 32, DDIM / 32, KSTEPS), 256, 0, stream>>>(W, Wt, DDIM, DDIM);
    // zt[b][t][d] = z[b][d][t]
    transpose_kernel<<<dim3(TLEN / 32, DDIM / 32, BSZ), 256, 0, stream>>>(z, zt, DDIM, TLEN);

    for (int k = 0; k < KSTEPS; ++k) {
        gemm_kernel<<<GEMM_BLOCKS, 256, 0, stream>>>(
            c, Wt + (size_t)k * DDIM * DDIM, bias + (size_t)k * DDIM, cst);
        score_kernel<<<SCORE_BLOCKS, 256, 0, stream>>>(
            zt, cst, neg + (size_t)k * BSZ * TLEN * NNEG,
            partials + (size_t)k * SCORE_BLOCKS * 2, k);
    }
    reduce_kernel<<<1, 32, 0, stream>>>(partials, out);
}